// AttentionDecoder_51539607607
// MI455X (gfx1250) — compile-verified
//
#include <hip/hip_runtime.h>
#include <hip/hip_bf16.h>
#include <math.h>

// Problem constants (from reference)
#define S_LEN 2000
#define NB    1024
#define EDIM  128
#define NHEAD 8
#define HDIM  16

#define CH     16              // s-rows per TDM tile (2000 = 125*16, no tail)
#define KSTR   132             // padded LDS row stride in floats (128 + 4)
#define NCHUNK (S_LEN / CH)    // 125

#define NEG_MIN (-3.402823466e38f)   // jnp.finfo(float32).min == -FLT_MAX

typedef __attribute__((ext_vector_type(2))) float v2f;
typedef __attribute__((ext_vector_type(8))) float v8f;
typedef __attribute__((ext_vector_type(4))) unsigned int v4u;
typedef __attribute__((ext_vector_type(8))) int v8i;
typedef __attribute__((ext_vector_type(4))) int v4i;

// ---------------------------------------------------------------------------
// TDM: load a CH x 128 f32 tile (row stride NB*EDIM elements in memory) into
// LDS at lds_byte_addr, padded to KSTR floats per row (pad_interval=128 DW,
// pad_amount=4 DW). D# per CDNA5 ISA.
// ---------------------------------------------------------------------------
__device__ __forceinline__ void tdm_load_tile(unsigned lds_byte_addr,
                                              const float* gptr)
{
    const unsigned long long ga = (unsigned long long)(size_t)gptr;
    // Group 0: count=1 | lds_addr | global_addr[56:0] | type=2
    const v4u g0 = { 1u,
                     lds_byte_addr,
                     (unsigned)(ga & 0xFFFFFFFFu),
                     (unsigned)((ga >> 32) & 0x1FFFFFFu) | (2u << 30) };
    // Group 1:
    //   [17:16] data_size=2 (4B) | [20] pad_enable | [24:22] pad_interval=6
    //   (128 DWORDs) | [31:25] pad_amount=3 (4 DWORDs)
    const unsigned e0 = (2u << 16) | (1u << 20) | (6u << 22) | (3u << 25);
    const v8i g1 = { (int)e0,
                     (int)(128u << 16),        // tensor_dim0 = 128 (bits 79:48)
                     (int)((unsigned)CH << 16),// tensor_dim1 = 16  (bits 111:80)
                     (int)(128u << 16),        // tile_dim0 = 128   (bits 127:112)
                     (int)CH,                  // tile_dim1 = 16, tile_dim2 = 0
                     (int)(NB * EDIM),         // tensor_dim0_stride = 131072
                     0, 0 };
    const v4i z4 = { 0, 0, 0, 0 };
#if defined(__clang_major__) && (__clang_major__ >= 23)
    const v8i z8 = { 0, 0, 0, 0, 0, 0, 0, 0 };
    __builtin_amdgcn_tensor_load_to_lds(g0, g1, z4, z4, z8, 0);
#else
    __builtin_amdgcn_tensor_load_to_lds(g0, g1, z4, z4, 0);
#endif
}

// ---------------------------------------------------------------------------
// Kernel 1: glimpse multi-head attention (L=1), one workgroup per n,
// wave w == head h. TDM double-buffers K/V tiles into LDS; flash-style
// online softmax so each of K,V is streamed exactly once.
// Lane mapping: r = lane&15 -> s-row in chunk, dh = lane>>4 -> 8-float slice.
// ---------------------------------------------------------------------------
__global__ __launch_bounds__(256) void glimpse_attn_kernel(
    const float* __restrict__ query,        // (1,N,E)
    const float* __restrict__ gK,           // (S,N,E)
    const float* __restrict__ gV,           // (S,N,E)
    const unsigned char* __restrict__ mask, // (N,1,S) bool
    float* __restrict__ glimpse)            // (N,E) workspace
{
    __shared__ float kb[2][CH * KSTR];      // 2 x 8448 B
    __shared__ float vb[2][CH * KSTR];      // 2 x 8448 B

    const int n    = blockIdx.x;
    const int tid  = threadIdx.x;
    const int h    = tid >> 5;              // wave id == head
    const int lane = tid & 31;
    const int r    = lane & 15;             // row within chunk
    const int dh   = lane >> 4;             // 0: dims 0-7, 1: dims 8-15

    // this lane's 8 query components (wave-uniform per half)
    const float* qp = query + (size_t)n * EDIM + h * HDIM + dh * 8;
    const float4 qa = *(const float4*)(qp + 0);
    const float4 qb = *(const float4*)(qp + 4);

    const unsigned char* mrow = mask + (size_t)n * S_LEN;
    const size_t rstride = (size_t)NB * EDIM;        // floats between s rows
    const float* kbase = gK + (size_t)n * EDIM;
    const float* vbase = gV + (size_t)n * EDIM;

    // prologue: DMA chunk 0 into buffer 0 (wave 0 owns the TDM queue)
    if (h == 0) {
        tdm_load_tile((unsigned)(size_t)&kb[0][0], kbase);
        tdm_load_tile((unsigned)(size_t)&vb[0][0], vbase);
    }

    float m = NEG_MIN;
    float l = 0.0f;
    float acc[8];
    #pragma unroll
    for (int j = 0; j < 8; ++j) acc[j] = 0.0f;

    for (int c = 0; c < NCHUNK; ++c) {
        const int buf = c & 1;
        if (h == 0) {
            if (c + 1 < NCHUNK) {
                const size_t off = (size_t)(c + 1) * CH * rstride;
                tdm_load_tile((unsigned)(size_t)&kb[buf ^ 1][0], kbase + off);
                tdm_load_tile((unsigned)(size_t)&vb[buf ^ 1][0], vbase + off);
                __builtin_amdgcn_s_wait_tensorcnt(2);  // chunk c done (in-order)
            } else {
                __builtin_amdgcn_s_wait_tensorcnt(0);
            }
        }
        __syncthreads();                    // tile c visible to all waves

        // score: q_h . K[s]  (pair of lanes shares one row, 8 floats each)
        const float* kr = &kb[buf][r * KSTR + h * HDIM + dh * 8];
        const float4 ka = *(const float4*)(kr + 0);
        const float4 kc = *(const float4*)(kr + 4);
        float part = qa.x*ka.x + qa.y*ka.y + qa.z*ka.z + qa.w*ka.w
                   + qb.x*kc.x + qb.y*kc.y + qb.z*kc.z + qb.w*kc.w;
        float d = part + __shfl_xor(part, 16, 32);     // full 16-dot in pair
        d *= 0.25f;                                     // 1/sqrt(16)
        const int s = c * CH + r;
        if (mrow[s]) d = NEG_MIN;

        // online softmax update (reductions stay inside each 16-lane half)
        float cmax = d;
        cmax = fmaxf(cmax, __shfl_xor(cmax, 1, 32));
        cmax = fmaxf(cmax, __shfl_xor(cmax, 2, 32));
        cmax = fmaxf(cmax, __shfl_xor(cmax, 4, 32));
        cmax = fmaxf(cmax, __shfl_xor(cmax, 8, 32));
        const float m_new = fmaxf(m, cmax);
        const float scf = __expf(m - m_new);
        const float p   = __expf(d - m_new);
        float csum = p;
        csum += __shfl_xor(csum, 1, 32);
        csum += __shfl_xor(csum, 2, 32);
        csum += __shfl_xor(csum, 4, 32);
        csum += __shfl_xor(csum, 8, 32);
        l = l * scf + csum;

        const float* vr = &vb[buf][r * KSTR + h * HDIM + dh * 8];
        const float4 va = *(const float4*)(vr + 0);
        const float4 vc = *(const float4*)(vr + 4);
        acc[0] = acc[0] * scf + p * va.x;
        acc[1] = acc[1] * scf + p * va.y;
        acc[2] = acc[2] * scf + p * va.z;
        acc[3] = acc[3] * scf + p * va.w;
        acc[4] = acc[4] * scf + p * vc.x;
        acc[5] = acc[5] * scf + p * vc.y;
        acc[6] = acc[6] * scf + p * vc.z;
        acc[7] = acc[7] * scf + p * vc.w;
        m = m_new;

        __syncthreads();                    // buffer free for next prefetch
    }

    // sum rows: reduce across the 16 lanes of each half
    #pragma unroll
    for (int off = 8; off >= 1; off >>= 1) {
        #pragma unroll
        for (int j = 0; j < 8; ++j)
            acc[j] += __shfl_xor(acc[j], off, 32);
    }

    if (r == 0) {
        const float invl = 1.0f / l;        // l already full-sum in every lane
        float* op = glimpse + (size_t)n * EDIM + h * HDIM + dh * 8;
        float4 o0 = { acc[0]*invl, acc[1]*invl, acc[2]*invl, acc[3]*invl };
        float4 o1 = { acc[4]*invl, acc[5]*invl, acc[6]*invl, acc[7]*invl };
        *(float4*)(op + 0) = o0;
        *(float4*)(op + 4) = o1;
    }
}

// ---------------------------------------------------------------------------
// Kernel 2: final_Q = glimpse @ W^T + b using V_WMMA_F32_16X16X4_F32.
// One wave per 16x16 output tile; K=128 in 32 WMMA steps of K=4.
// ---------------------------------------------------------------------------
__global__ __launch_bounds__(32) void proj_wmma_kernel(
    const float* __restrict__ A,     // glimpse (1024,128)
    const float* __restrict__ W,     // (128,128)  (y = x @ W^T + b)
    const float* __restrict__ bias,  // (128)
    float* __restrict__ out)         // final_Q (1024,128)
{
    const int lane = threadIdx.x;
    const int mn   = lane & 15;
    const int koff = (lane >> 4) << 1;      // 0 or 2
    const int row  = blockIdx.x * 16 + mn;  // A row for A-frag
    const int col  = blockIdx.y * 16 + mn;  // B col for B-frag / D col

    v8f acc = {0.f, 0.f, 0.f, 0.f, 0.f, 0.f, 0.f, 0.f};

    #pragma unroll 4
    for (int k0 = 0; k0 < EDIM; k0 += 4) {
        const v2f a  = *(const v2f*)(A + (size_t)row * EDIM + k0 + koff);
        const v2f bb = *(const v2f*)(W + (size_t)col * EDIM + k0 + koff);
        acc = __builtin_amdgcn_wmma_f32_16x16x4_f32(
                  false, a, false, bb, (short)0, acc, false, false);
    }

    const float bv = bias[col];
    const int rbase = blockIdx.x * 16 + ((lane >> 4) << 3);  // +8 for hi half
    #pragma unroll
    for (int rr = 0; rr < 8; ++rr) {
        out[(size_t)(rbase + rr) * EDIM + col] = acc[rr] + bv;
    }
}

// ---------------------------------------------------------------------------
// Kernel 3: logits = tanh(final_Q . logit_K / sqrt(128))*10, mask, log_softmax.
// One WG per n; logit_K[n] is contiguous, half-wave cooperative dot products.
// ---------------------------------------------------------------------------
__global__ __launch_bounds__(256) void logits_kernel(
    const float* __restrict__ fq,           // (N,128)
    const float* __restrict__ lK,           // (N,S,E)
    const unsigned char* __restrict__ mask, // (N,1,S)
    float* __restrict__ out)                // (N,S)
{
    __shared__ float lg[S_LEN];             // 8000 B
    __shared__ float red[8];

    const int n    = blockIdx.x;
    const int tid  = threadIdx.x;
    const int wave = tid >> 5;
    const int lane = tid & 31;
    const int half = lane >> 4;
    const int l16  = lane & 15;

    const float* qp = fq + (size_t)n * EDIM + l16 * 8;
    const float4 q0 = *(const float4*)(qp + 0);
    const float4 q1 = *(const float4*)(qp + 4);

    const float* kbase = lK + (size_t)n * S_LEN * EDIM;
    const unsigned char* mrow = mask + (size_t)n * S_LEN;
    const float scale = 0.088388347648318447f;  // 1/sqrt(128)

    for (int s0 = 0; s0 < S_LEN; s0 += 16) {    // 2000 % 16 == 0
        const int s = s0 + wave * 2 + half;
        const float* kp = kbase + (size_t)s * EDIM + l16 * 8;
        const float4 k0 = *(const float4*)(kp + 0);
        const float4 k1 = *(const float4*)(kp + 4);
        float d = q0.x*k0.x + q0.y*k0.y + q0.z*k0.z + q0.w*k0.w
                + q1.x*k1.x + q1.y*k1.y + q1.z*k1.z + q1.w*k1.w;
        d += __shfl_xor(d, 1, 32);
        d += __shfl_xor(d, 2, 32);
        d += __shfl_xor(d, 4, 32);
        d += __shfl_xor(d, 8, 32);
        float v = tanhf(d * scale) * 10.0f;
        if (mrow[s]) v = NEG_MIN;
        if (l16 == 0) lg[s] = v;
    }
    __syncthreads();

    float m = NEG_MIN;
    for (int s = tid; s < S_LEN; s += 256) m = fmaxf(m, lg[s]);
    #pragma unroll
    for (int off = 16; off >= 1; off >>= 1)
        m = fmaxf(m, __shfl_xor(m, off, 32));
    if (lane == 0) red[wave] = m;
    __syncthreads();
    m = red[0];
    #pragma unroll
    for (int w = 1; w < 8; ++w) m = fmaxf(m, red[w]);

    float l = 0.0f;
    for (int s = tid; s < S_LEN; s += 256) l += __expf(lg[s] - m);
    #pragma unroll
    for (int off = 16; off >= 1; off >>= 1)
        l += __shfl_xor(l, off, 32);
    __syncthreads();                    // all done reading red[] (max)
    if (lane == 0) red[wave] = l;
    __syncthreads();
    l = 0.0f;
    #pragma unroll
    for (int w = 0; w < 8; ++w) l += red[w];

    const float logZ = m + __logf(l);
    float* orow = out + (size_t)n * S_LEN;
    for (int s = tid; s < S_LEN; s += 256) orow[s] = lg[s] - logZ;
}

// ---------------------------------------------------------------------------
extern "C" void kernel_launch(void* const* d_in, const int* in_sizes, int n_in,
                              void* d_out, int out_size, void* d_ws, size_t ws_size,
                              hipStream_t stream) {
    (void)in_sizes; (void)n_in; (void)out_size; (void)ws_size;

    const float*         query = (const float*)d_in[0];         // (1,N,E)
    const float*         gK    = (const float*)d_in[1];         // (S,N,E)
    const float*         gV    = (const float*)d_in[2];         // (S,N,E)
    const float*         lK    = (const float*)d_in[3];         // (N,S,E)
    const unsigned char* mask  = (const unsigned char*)d_in[4]; // (N,1,S) bool
    const float*         W     = (const float*)d_in[5];         // (E,E)
    const float*         bias  = (const float*)d_in[6];         // (E)
    float*               out   = (float*)d_out;                 // (N,S)

    float* glimpse = (float*)d_ws;                 // N*E floats (512 KB)
    float* finalQ  = glimpse + (size_t)NB * EDIM;  // N*E floats (512 KB)

    glimpse_attn_kernel<<<NB, 256, 0, stream>>>(query, gK, gV, mask, glimpse);

    dim3 g2(NB / 16, EDIM / 16);                   // 64 x 8 tiles, 1 wave each
    proj_wmma_kernel<<<g2, 32, 0, stream>>>(glimpse, W, bias, finalQ);

    logits_kernel<<<NB, 256, 0, stream>>>(finalQ, lK, mask, out);
}